// _SAGE_59133109731936
// MI455X (gfx1250) — compile-verified
//
#include <hip/hip_runtime.h>

// fp32 WMMA fragment types (wave32): A/B = 16x4 / 4x16 f32 -> 2 VGPRs, C/D = 16x16 f32 -> 8 VGPRs
typedef __attribute__((ext_vector_type(2))) float v2f;
typedef __attribute__((ext_vector_type(8))) float v8f;

#define FDIM 128
#define LDSROW 132  // padded row stride (floats): lanes hit distinct LDS banks on ds_load_b64

// ---------------------------------------------------------------- utilities
__global__ __launch_bounds__(256) void zero_f32(float* __restrict__ p, long long n) {
    long long i4 = ((long long)blockIdx.x * blockDim.x + threadIdx.x) * 4;
    if (i4 + 3 < n) {
        *(float4*)(p + i4) = make_float4(0.f, 0.f, 0.f, 0.f);
    } else {
        for (long long j = i4; j < n; ++j) p[j] = 0.f;
    }
}

__global__ __launch_bounds__(256) void copy_f32_vec4(float* __restrict__ d,
                                                     const float* __restrict__ s,
                                                     long long n4) {
    long long i = (long long)blockIdx.x * blockDim.x + threadIdx.x;
    if (i < n4) ((float4*)d)[i] = ((const float4*)s)[i];
}

// ---------------------------------------------------------------- scatter (segment sum)
// One wave32 per edge: wave reads the 128-float feature row of src coalesced,
// each lane atomically adds 4 floats into s[dst]. Lane group c==0 bumps count.
__global__ __launch_bounds__(256) void sage_scatter(const float* __restrict__ xin,
                                                    const int* __restrict__ src,
                                                    const int* __restrict__ dst,
                                                    float* __restrict__ ssum,
                                                    float* __restrict__ cnt,
                                                    int n_edges) {
    long long t = (long long)blockIdx.x * blockDim.x + threadIdx.x;
    int e = (int)(t >> 5);
    if (e >= n_edges) return;
    int c = ((int)t & 31) << 2;  // 0..124
    int sn = src[e];
    int dn = dst[e];
    float4 v = *(const float4*)&xin[(long long)sn * FDIM + c];
    float* p = &ssum[(long long)dn * FDIM + c];
    atomicAdd(p + 0, v.x);
    atomicAdd(p + 1, v.y);
    atomicAdd(p + 2, v.z);
    atomicAdd(p + 3, v.w);
    if (cnt != nullptr && c == 0) atomicAdd(&cnt[dn], 1.0f);
}

// ---------------------------------------------------------------- fused SAGE GEMM
// out_c[m,:] = (ssum[m,:]/max(cnt[m],1)) @ wl^T + bl + xin[m,:] @ wr^T
// optional: out_relu = max(out_c, 0)
// Block = 128 threads (4 waves), handles 16 node rows x 128 output cols.
// Wave w owns output column tiles [32w,32w+16) and [32w+16,32w+32).
__global__ __launch_bounds__(128) void sage_gemm(const float* __restrict__ xin,
                                                 const float* __restrict__ ssum,
                                                 const float* __restrict__ cnt,
                                                 const float* __restrict__ wl,
                                                 const float* __restrict__ bl,
                                                 const float* __restrict__ wr,
                                                 float* __restrict__ out_c,
                                                 float* __restrict__ out_relu,
                                                 int n_nodes) {
    __shared__ float lds_agg[16 * LDSROW];
    __shared__ float lds_x[16 * LDSROW];

    const int tid = threadIdx.x;
    const int node0 = blockIdx.x * 16;

    // Cooperative stage of 16 rows (x and mean-normalized sums) into LDS.
    // 2 x 2048 floats, 128 threads -> 4 float4 per thread per matrix, coalesced.
    for (int it = 0; it < 4; ++it) {
        int v = it * 128 + tid;        // 0..511 vector index
        int row = v >> 5;              // 0..15
        int col = (v & 31) << 2;       // 0..124
        int node = node0 + row;
        float4 xv = make_float4(0.f, 0.f, 0.f, 0.f);
        float4 sv = make_float4(0.f, 0.f, 0.f, 0.f);
        if (node < n_nodes) {
            xv = *(const float4*)&xin[(long long)node * FDIM + col];
            sv = *(const float4*)&ssum[(long long)node * FDIM + col];
            float invc = 1.0f / fmaxf(cnt[node], 1.0f);
            sv.x *= invc; sv.y *= invc; sv.z *= invc; sv.w *= invc;
        }
        *(float4*)&lds_x[row * LDSROW + col] = xv;
        *(float4*)&lds_agg[row * LDSROW + col] = sv;
    }
    __syncthreads();

    const int wave = tid >> 5;
    const int lane = tid & 31;
    const int mrow = lane & 15;            // A row index (M)
    const int khalf = (lane >> 4) << 1;    // lanes 16-31 hold K+2,K+3
    const int ncol0 = wave * 32 + (lane & 15);
    const int ncol1 = ncol0 + 16;

    // D/C init with bias broadcast (every element of a column gets bl[n]).
    v8f acc0, acc1;
    float bias0 = bl[ncol0];
    float bias1 = bl[ncol1];
#pragma unroll
    for (int i = 0; i < 8; ++i) { acc0[i] = bias0; acc1[i] = bias1; }

    // B[k][n] = W[n][k]: each lane reads 2 consecutive K from weight row n.
    const float* wl0 = wl + (long long)ncol0 * FDIM;
    const float* wl1 = wl + (long long)ncol1 * FDIM;
    const float* wr0 = wr + (long long)ncol0 * FDIM;
    const float* wr1 = wr + (long long)ncol1 * FDIM;
    const float* la = &lds_agg[mrow * LDSROW];
    const float* lx = &lds_x[mrow * LDSROW];

#pragma unroll 4
    for (int k0 = 0; k0 < FDIM; k0 += 4) {
        int kk = k0 + khalf;
        v2f a_agg = *(const v2f*)(la + kk);    // ds_load_b64, bank-conflict-free
        v2f a_x   = *(const v2f*)(lx + kk);
        v2f bL0 = *(const v2f*)(wl0 + kk);     // global_load_b64 (L2-resident weights)
        v2f bR0 = *(const v2f*)(wr0 + kk);
        v2f bL1 = *(const v2f*)(wl1 + kk);
        v2f bR1 = *(const v2f*)(wr1 + kk);
        // D = A*B + C, fp32 matrix pipe
        acc0 = __builtin_amdgcn_wmma_f32_16x16x4_f32(false, a_agg, false, bL0, (short)0, acc0, false, false);
        acc0 = __builtin_amdgcn_wmma_f32_16x16x4_f32(false, a_x,   false, bR0, (short)0, acc0, false, false);
        acc1 = __builtin_amdgcn_wmma_f32_16x16x4_f32(false, a_agg, false, bL1, (short)0, acc1, false, false);
        acc1 = __builtin_amdgcn_wmma_f32_16x16x4_f32(false, a_x,   false, bR1, (short)0, acc1, false, false);
    }

    // D layout: VGPR i -> M = i (lanes 0-15) / i+8 (lanes 16-31); N = lane&15 within tile.
    const int mbase = (lane >> 4) << 3;
#pragma unroll
    for (int i = 0; i < 8; ++i) {
        int node = node0 + mbase + i;
        if (node < n_nodes) {
            float c0 = acc0[i];
            float c1 = acc1[i];
            long long base = (long long)node * FDIM;
            out_c[base + ncol0] = c0;
            out_c[base + ncol1] = c1;
            if (out_relu != nullptr) {
                out_relu[base + ncol0] = fmaxf(c0, 0.0f);
                out_relu[base + ncol1] = fmaxf(c1, 0.0f);
            }
        }
    }
}

// ---------------------------------------------------------------- launcher
extern "C" void kernel_launch(void* const* d_in, const int* in_sizes, int n_in,
                              void* d_out, int out_size, void* d_ws, size_t ws_size,
                              hipStream_t stream) {
    const float* x   = (const float*)d_in[0];
    const int*   ei  = (const int*)d_in[1];
    const float* w1l = (const float*)d_in[2];
    const float* b1l = (const float*)d_in[3];
    const float* w1r = (const float*)d_in[4];
    const float* w2l = (const float*)d_in[5];
    const float* b2l = (const float*)d_in[6];
    const float* w2r = (const float*)d_in[7];

    const int n_nodes = in_sizes[0] / FDIM;
    const int n_edges = in_sizes[1] / 2;
    const int* src = ei;
    const int* dst = ei + n_edges;

    float* out    = (float*)d_out;
    float* out_x  = out;
    float* out_c1 = out + (long long)n_nodes * FDIM;
    float* out_c2 = out + 2LL * (long long)n_nodes * FDIM;

    // workspace: segment sums [N*F], counts [N], relu(c1) [N*F]
    float* ssum = (float*)d_ws;
    float* cnt  = ssum + (long long)n_nodes * FDIM;
    float* a1   = ssum + (((long long)n_nodes * FDIM + n_nodes + 3LL) / 4LL) * 4LL;

    const long long nf = (long long)n_nodes * FDIM;
    const int gemm_blocks = (n_nodes + 15) / 16;
    const long long scat_threads = (long long)n_edges * 32;
    const int scat_blocks = (int)((scat_threads + 255) / 256);

    // ---- layer 1 ----
    {
        long long zc = nf + n_nodes;  // sums + counts contiguous
        int blocks = (int)(((zc + 3) / 4 + 255) / 256);
        zero_f32<<<blocks, 256, 0, stream>>>(ssum, zc);
    }
    {
        long long n4 = nf / 4;
        int blocks = (int)((n4 + 255) / 256);
        copy_f32_vec4<<<blocks, 256, 0, stream>>>(out_x, x, n4);
    }
    sage_scatter<<<scat_blocks, 256, 0, stream>>>(x, src, dst, ssum, cnt, n_edges);
    sage_gemm<<<gemm_blocks, 128, 0, stream>>>(x, ssum, cnt, w1l, b1l, w1r,
                                               out_c1, a1, n_nodes);

    // ---- layer 2 (counts unchanged; re-zero sums only) ----
    {
        int blocks = (int)(((nf + 3) / 4 + 255) / 256);
        zero_f32<<<blocks, 256, 0, stream>>>(ssum, nf);
    }
    sage_scatter<<<scat_blocks, 256, 0, stream>>>(a1, src, dst, ssum, nullptr, n_edges);
    sage_gemm<<<gemm_blocks, 128, 0, stream>>>(a1, ssum, cnt, w2l, b2l, w2r,
                                               out_c2, nullptr, n_nodes);
}